// GCNNet_56977036149347
// MI455X (gfx1250) — compile-verified
//
#include <hip/hip_runtime.h>
#include <math.h>

#define N_NODES 100000
#define N_EDGES 600000

typedef float v2f __attribute__((ext_vector_type(2)));
typedef float v8f __attribute__((ext_vector_type(8)));

// ---------------------------------------------------------------------------
// Utility fills
// ---------------------------------------------------------------------------
__global__ void fill1_kernel(float* __restrict__ p, float v, int n) {
    int i = blockIdx.x * blockDim.x + threadIdx.x;
    if (i < n) p[i] = v;
}

__global__ void zero4_kernel(float4* __restrict__ p, long long n4) {
    long long i = (long long)blockIdx.x * blockDim.x + threadIdx.x;
    if (i < n4) p[i] = make_float4(0.f, 0.f, 0.f, 0.f);
}

// deg[dst] += 1 for every real edge (self-loop pre-counted by fill of 1.0)
__global__ void degree_kernel(const int* __restrict__ ei, float* __restrict__ deg, int E) {
    int e = blockIdx.x * blockDim.x + threadIdx.x;
    if (e < E) atomicAdd(&deg[ei[E + e]], 1.0f);
}

__global__ void rsqrt_kernel(float* __restrict__ d, int n) {
    int i = blockIdx.x * blockDim.x + threadIdx.x;
    if (i < n) d[i] = rsqrtf(d[i]);   // deg >= 1 always (self-loops)
}

// ---------------------------------------------------------------------------
// Row-scaled GEMM via V_WMMA_F32_16X16X4_F32:  out[i,:] = (A[i,:] @ W) * dinv[i]
// A: [nrows x K] row-major, W: [K x NCOLS] row-major. One block = 16-row tile,
// one wave per 16-column tile (NCOLS/16 waves). nrows % 16 == 0 (100000).
// ---------------------------------------------------------------------------
template <int K, int NCOLS>
__global__ void gemm_scale_kernel(const float* __restrict__ A,
                                  const float* __restrict__ W,
                                  const float* __restrict__ dinv,
                                  float* __restrict__ out) {
    const int lane = threadIdx.x & 31;
    const int wave = threadIdx.x >> 5;
    const int m0   = blockIdx.x * 16;
    const int n0   = wave * 16;
    const int half = lane >> 4;      // 0: K-pair {0,1}, 1: K-pair {2,3}
    const int l16  = lane & 15;

    const float* Arow = A + (size_t)(m0 + l16) * K;  // A row this lane feeds
    v8f c = {0.f, 0.f, 0.f, 0.f, 0.f, 0.f, 0.f, 0.f};

#pragma unroll
    for (int k0 = 0; k0 < K; k0 += 4) {
        const int ka = k0 + 2 * half;
        v2f a;
        a.x = Arow[ka];
        a.y = Arow[ka + 1];
        v2f b;
        b.x = W[(size_t)ka * NCOLS + n0 + l16];
        b.y = W[(size_t)(ka + 1) * NCOLS + n0 + l16];
        c = __builtin_amdgcn_wmma_f32_16x16x4_f32(false, a, false, b,
                                                  (short)0, c, false, false);
    }

    // D layout: lane holds D[m0 + r + 8*half][n0 + l16], r = 0..7
#pragma unroll
    for (int r = 0; r < 8; ++r) {
        const int gr = m0 + r + 8 * half;
        out[(size_t)gr * NCOLS + n0 + l16] = c[r] * dinv[gr];
    }
}

// ---------------------------------------------------------------------------
// Edge scatter: acc[dst,:] += xs[src,:]   (float4 per thread, f32 L2 atomics)
// ---------------------------------------------------------------------------
template <int NCOLS>
__global__ void scatter_kernel(const int* __restrict__ ei, int E,
                               const float* __restrict__ xs,
                               float* __restrict__ acc) {
    constexpr int CH = NCOLS / 4;
    long long tid = (long long)blockIdx.x * blockDim.x + threadIdx.x;
    if (tid >= (long long)E * CH) return;
    const int e = (int)(tid / CH);
    const int c = (int)(tid % CH) * 4;
    const int s = ei[e];
    const int d = ei[E + e];
    const float4 v = *(const float4*)(xs + (size_t)s * NCOLS + c);
    float* p = acc + (size_t)d * NCOLS + c;
    atomicAdd(p + 0, v.x);
    atomicAdd(p + 1, v.y);
    atomicAdd(p + 2, v.z);
    atomicAdd(p + 3, v.w);
}

// ---------------------------------------------------------------------------
// Finalize: h = relu(dinv[n] * (acc + xs) + bias)   (in place into acc)
// ---------------------------------------------------------------------------
template <int NCOLS>
__global__ void fin_relu_kernel(float* __restrict__ acc,
                                const float* __restrict__ xs,
                                const float* __restrict__ dinv,
                                const float* __restrict__ bias, int nrows) {
    constexpr int CH = NCOLS / 4;
    long long tid = (long long)blockIdx.x * blockDim.x + threadIdx.x;
    if (tid >= (long long)nrows * CH) return;
    const int n = (int)(tid / CH);
    const int c = (int)(tid % CH) * 4;
    const float di = dinv[n];
    const size_t off = (size_t)n * NCOLS + c;
    float4 a = *(float4*)(acc + off);
    const float4 x = *(const float4*)(xs + off);
    a.x = fmaxf(fmaf(di, a.x + x.x, bias[c + 0]), 0.f);
    a.y = fmaxf(fmaf(di, a.y + x.y, bias[c + 1]), 0.f);
    a.z = fmaxf(fmaf(di, a.z + x.z, bias[c + 2]), 0.f);
    a.w = fmaxf(fmaf(di, a.w + x.w, bias[c + 3]), 0.f);
    *(float4*)(acc + off) = a;
}

// ---------------------------------------------------------------------------
// Head: out[n] = sigmoid( gelu(h2[n,:] @ Wh1 + bh1) @ Wh2 + bh2 )
// ---------------------------------------------------------------------------
__global__ void head_kernel(const float* __restrict__ h2,
                            const float* __restrict__ Wh1,
                            const float* __restrict__ bh1,
                            const float* __restrict__ Wh2,
                            const float* __restrict__ bh2,
                            float* __restrict__ out, int n) {
    __shared__ float sW1[64 * 16];
    __shared__ float sb1[16];
    __shared__ float sW2[16];
    __shared__ float sb2;
    for (int i = threadIdx.x; i < 64 * 16; i += blockDim.x) sW1[i] = Wh1[i];
    if (threadIdx.x < 16) {
        sb1[threadIdx.x] = bh1[threadIdx.x];
        sW2[threadIdx.x] = Wh2[threadIdx.x];
    }
    if (threadIdx.x == 0) sb2 = bh2[0];
    __syncthreads();

    const int node = blockIdx.x * blockDim.x + threadIdx.x;
    if (node >= n) return;

    float accv[16];
#pragma unroll
    for (int t = 0; t < 16; ++t) accv[t] = sb1[t];

    const float* row = h2 + (size_t)node * 64;
    for (int j = 0; j < 64; ++j) {
        const float v = row[j];
#pragma unroll
        for (int t = 0; t < 16; ++t) accv[t] = fmaf(v, sW1[j * 16 + t], accv[t]);
    }

    float s = sb2;
#pragma unroll
    for (int t = 0; t < 16; ++t) {
        const float a = accv[t];
        const float g = 0.5f * a * (1.0f + erff(a * 0.70710678118654752440f)); // exact gelu
        s = fmaf(g, sW2[t], s);
    }
    out[node] = 1.0f / (1.0f + expf(-s));
}

// ---------------------------------------------------------------------------
// Launch sequence
// ---------------------------------------------------------------------------
extern "C" void kernel_launch(void* const* d_in, const int* in_sizes, int n_in,
                              void* d_out, int out_size, void* d_ws, size_t ws_size,
                              hipStream_t stream) {
    const float* x   = (const float*)d_in[0];
    const int*   ei  = (const int*)d_in[1];    // [2, E]: src rows then dst rows
    const float* W1  = (const float*)d_in[2];
    const float* b1  = (const float*)d_in[3];
    const float* W2  = (const float*)d_in[4];
    const float* b2  = (const float*)d_in[5];
    const float* Wh1 = (const float*)d_in[6];
    const float* bh1 = (const float*)d_in[7];
    const float* Wh2 = (const float*)d_in[8];
    const float* bh2 = (const float*)d_in[9];
    float* out = (float*)d_out;

    const int N = N_NODES;
    const int E = N_EDGES;

    // Workspace layout (floats): dinv[N] | bufA[N*128] | bufB[N*128]
    float* dinv = (float*)d_ws;
    float* bufA = dinv + N;            // xs1, later xs2
    float* bufB = bufA + (size_t)N * 128;  // acc1 -> h1 -> acc2 -> h2

    const int TB = 256;

    // deg init (1.0 self-loop) + zero acc1
    fill1_kernel<<<(N + TB - 1) / TB, TB, 0, stream>>>(dinv, 1.0f, N);
    {
        long long n4 = (long long)N * 128 / 4;
        zero4_kernel<<<(unsigned)((n4 + TB - 1) / TB), TB, 0, stream>>>((float4*)bufB, n4);
    }
    degree_kernel<<<(E + TB - 1) / TB, TB, 0, stream>>>(ei, dinv, E);
    rsqrt_kernel<<<(N + TB - 1) / TB, TB, 0, stream>>>(dinv, N);

    // ---- Layer 1: xs1 = (x@W1)*dinv ; scatter ; h1 = relu(dinv*(acc+xs)+b1)
    gemm_scale_kernel<128, 128><<<N / 16, 256, 0, stream>>>(x, W1, dinv, bufA);
    {
        long long t = (long long)E * 32;
        scatter_kernel<128><<<(unsigned)((t + TB - 1) / TB), TB, 0, stream>>>(ei, E, bufA, bufB);
        long long f = (long long)N * 32;
        fin_relu_kernel<128><<<(unsigned)((f + TB - 1) / TB), TB, 0, stream>>>(bufB, bufA, dinv, b1, N);
    }

    // ---- Layer 2: xs2 = (h1@W2)*dinv ; re-zero acc ; scatter ; h2 = relu(...)
    gemm_scale_kernel<128, 64><<<N / 16, 128, 0, stream>>>(bufB, W2, dinv, bufA);
    {
        long long n4 = (long long)N * 64 / 4;
        zero4_kernel<<<(unsigned)((n4 + TB - 1) / TB), TB, 0, stream>>>((float4*)bufB, n4);
        long long t = (long long)E * 16;
        scatter_kernel<64><<<(unsigned)((t + TB - 1) / TB), TB, 0, stream>>>(ei, E, bufA, bufB);
        long long f = (long long)N * 16;
        fin_relu_kernel<64><<<(unsigned)((f + TB - 1) / TB), TB, 0, stream>>>(bufB, bufA, dinv, b2, N);
    }

    // ---- Head: gelu(h2@Wh1+bh1) @ Wh2 + bh2 -> sigmoid
    head_kernel<<<(N + TB - 1) / TB, TB, 0, stream>>>(bufB, Wh1, bh1, Wh2, bh2, out, N);
}